// LinearTransformerBlock_5153960755280
// MI455X (gfx1250) — compile-verified
//
#include <hip/hip_runtime.h>
#include <hip/hip_bf16.h>

typedef __attribute__((ext_vector_type(2))) float v2f;
typedef __attribute__((ext_vector_type(8))) float v8f;

#define NB   2
#define SEQ  512
#define DM   512          // model dim
#define NH   8
#define HD   64
#define HID  512          // NH*HD
#define FFN  2048
#define BSR  (NB*SEQ)     // 1024 rows

// ---------------- RMSNorm: one 256-thread block per row ----------------
__global__ void rmsnorm_kernel(const float* __restrict__ x,
                               const float* __restrict__ scale,
                               float* __restrict__ out, int ncols) {
  __shared__ float red[256];
  int row = blockIdx.x;
  const float* xr = x + (size_t)row * ncols;
  float ss = 0.f;
  for (int c = threadIdx.x; c < ncols; c += 256) { float v = xr[c]; ss += v * v; }
  red[threadIdx.x] = ss;
  __syncthreads();
  for (int off = 128; off > 0; off >>= 1) {
    if (threadIdx.x < off) red[threadIdx.x] += red[threadIdx.x + off];
    __syncthreads();
  }
  float inv = rsqrtf(red[0] / (float)ncols + 1e-6f);
  float* orow = out + (size_t)row * ncols;
  for (int c = threadIdx.x; c < ncols; c += 256) orow[c] = xr[c] * inv * scale[c];
}

// ---------------- FP32 GEMM via V_WMMA_F32_16X16X4_F32 ----------------
// C[M,N] = act(X[M,K] @ W[K,N] + bias) + resid
// 8 waves/block; each wave computes a 32x32 tile as 2x2 WMMA tiles
// (4 v8f accumulators) so each A/B fragment pair feeds 2 WMMAs.
// act: 0 = none, 1 = relu
__global__ void gemm_wmma_f32(const float* __restrict__ X, const float* __restrict__ W,
                              const float* __restrict__ bias, const float* __restrict__ resid,
                              float* __restrict__ out, int M, int N, int K, int act) {
  int wave = threadIdx.x >> 5;
  int lane = threadIdx.x & 31;
  int tile = blockIdx.x * 8 + wave;       // 32x32 macro-tile index
  int ntn  = N >> 5;
  int tm   = tile / ntn;
  int tn   = tile - tm * ntn;
  if (tm >= (M >> 5)) return;             // uniform per wave

  int r0   = tm * 32;
  int c0   = tn * 32;
  int lm   = lane & 15;
  int koff = (lane >> 4) * 2;             // lanes 16..31 carry K+2,K+3

  const float* Xp0 = X + (size_t)(r0 + lm) * K + koff;        // A rows 0..15
  const float* Xp1 = X + (size_t)(r0 + 16 + lm) * K + koff;   // A rows 16..31
  const float* Wp0 = W + (size_t)koff * N + (c0 + lm);        // B cols 0..15
  const float* Wp1 = W + (size_t)koff * N + (c0 + 16 + lm);   // B cols 16..31

  v8f acc[2][2] = {};
  for (int k = 0; k < K; k += 4) {
    v2f a0 = *(const v2f*)(Xp0 + k);
    v2f a1 = *(const v2f*)(Xp1 + k);
    v2f b0, b1;
    b0.x = Wp0[(size_t)k * N];
    b0.y = Wp0[(size_t)(k + 1) * N];
    b1.x = Wp1[(size_t)k * N];
    b1.y = Wp1[(size_t)(k + 1) * N];
    acc[0][0] = __builtin_amdgcn_wmma_f32_16x16x4_f32(false, a0, false, b0, (short)0, acc[0][0], false, false);
    acc[0][1] = __builtin_amdgcn_wmma_f32_16x16x4_f32(false, a0, false, b1, (short)0, acc[0][1], false, false);
    acc[1][0] = __builtin_amdgcn_wmma_f32_16x16x4_f32(false, a1, false, b0, (short)0, acc[1][0], false, false);
    acc[1][1] = __builtin_amdgcn_wmma_f32_16x16x4_f32(false, a1, false, b1, (short)0, acc[1][1], false, false);
  }

  int rhalf = (lane >> 4) << 3;           // upper half-wave holds rows M+8..M+15
#pragma unroll
  for (int mi = 0; mi < 2; ++mi) {
#pragma unroll
    for (int ni = 0; ni < 2; ++ni) {
      int rbase = r0 + mi * 16 + rhalf;
      int col   = c0 + ni * 16 + lm;
      float bval = bias ? bias[col] : 0.f;
#pragma unroll
      for (int r = 0; r < 8; ++r) {
        float vv = acc[mi][ni][r] + bval;
        if (act == 1) vv = fmaxf(vv, 0.f);
        size_t idx = (size_t)(rbase + r) * N + col;
        if (resid) vv += resid[idx];
        out[idx] = vv;
      }
    }
  }
}

// ---------------- Sequential linear-attention scan ----------------
// One block per (b,h); 64x64 state in registers (16 per thread).
// S_s = (1-mask_s)*S_{s-1} + v_s (x) k_s ;  attn_s = S_s @ q_s
__global__ void scan_kernel(const float* __restrict__ q, const float* __restrict__ k,
                            const float* __restrict__ v, const float* __restrict__ mask,
                            const float* __restrict__ carry, float* __restrict__ attn,
                            float* __restrict__ carry_out) {
  __shared__ float ksh[HD], vsh[HD], qsh[HD], red[4 * HD];
  int bh = blockIdx.x;            // b*NH + h
  int b  = bh / NH;
  int h  = bh - b * NH;
  int tid = threadIdx.x;          // 256 threads
  int i   = tid & 63;             // state row (v index)
  int jg  = tid >> 6;             // column group 0..3
  int j0  = jg * 16;

  float st[16];
  const float* cp = carry + ((size_t)bh * HD + i) * HD + j0;
#pragma unroll
  for (int jj = 0; jj < 16; ++jj) st[jj] = cp[jj];

  for (int s = 0; s < SEQ; ++s) {
    size_t base = ((size_t)(b * SEQ + s) * HID) + (size_t)h * HD;
    if (tid < 64)       ksh[tid]        = k[base + tid];
    else if (tid < 128) vsh[tid - 64]   = v[base + tid - 64];
    else if (tid < 192) qsh[tid - 128]  = q[base + tid - 128];
    __syncthreads();

    float c  = 1.0f - mask[b * SEQ + s];
    float vi = vsh[i];
    float acc = 0.f;
#pragma unroll
    for (int jj = 0; jj < 16; ++jj) {
      st[jj] = st[jj] * c + vi * ksh[j0 + jj];
      acc   += st[jj] * qsh[j0 + jj];
    }
    red[jg * 64 + i] = acc;
    __syncthreads();
    if (tid < 64)
      attn[base + tid] = red[tid] + red[64 + tid] + red[128 + tid] + red[192 + tid];
    __syncthreads();
  }

  float* co = carry_out + ((size_t)bh * HD + i) * HD + j0;
#pragma unroll
  for (int jj = 0; jj < 16; ++jj) co[jj] = st[jj];
}

// ---------------- SiLU gate (in place into g) ----------------
__global__ void silu_mul_kernel(float* __restrict__ g, const float* __restrict__ u, int n) {
  int idx = blockIdx.x * 256 + threadIdx.x;
  if (idx < n) {
    float gv = g[idx];
    g[idx] = gv / (1.f + expf(-gv)) * u[idx];
  }
}

extern "C" void kernel_launch(void* const* d_in, const int* in_sizes, int n_in,
                              void* d_out, int out_size, void* d_ws, size_t ws_size,
                              hipStream_t stream) {
  const float* inputs  = (const float*)d_in[0];
  const float* mask    = (const float*)d_in[1];
  const float* carry   = (const float*)d_in[2];
  const float* ln1     = (const float*)d_in[3];
  const float* Wq      = (const float*)d_in[4];
  const float* Wk      = (const float*)d_in[5];
  const float* Wv      = (const float*)d_in[6];
  const float* attn_ln = (const float*)d_in[7];
  const float* Wo      = (const float*)d_in[8];
  const float* bo      = (const float*)d_in[9];
  const float* ln2     = (const float*)d_in[10];
  const float* Wg      = (const float*)d_in[11];
  const float* Wu      = (const float*)d_in[12];
  const float* Wd      = (const float*)d_in[13];

  float* out_carry = (float*)d_out;                       // B*H*HD*HD = 65536
  float* out_x     = (float*)d_out + NB * NH * HD * HD;   // B*S*D

  // workspace carve-up (floats)
  float* ws    = (float*)d_ws;
  float* bufA  = ws;                        // BSR*DM : norm buffer (reused 3x)
  float* qbuf  = bufA + (size_t)BSR * DM;   // BSR*HID
  float* kbuf  = qbuf + (size_t)BSR * HID;
  float* vbuf  = kbuf + (size_t)BSR * HID;
  float* abuf  = vbuf + (size_t)BSR * HID;  // attn
  float* x1    = abuf + (size_t)BSR * HID;  // post-attn residual
  float* gbuf  = x1   + (size_t)BSR * DM;   // BSR*FFN
  float* ubuf  = gbuf + (size_t)BSR * FFN;  // BSR*FFN

  // 32x32 macro-tiles, 8 waves (tiles) per 256-thread block
  const int blocks512 = (BSR / 32) * (DM / 32) / 8;    // 64
  const int blocksFFN = (BSR / 32) * (FFN / 32) / 8;   // 256

  // 1) x = rmsnorm(inputs, ln1)
  rmsnorm_kernel<<<BSR, 256, 0, stream>>>(inputs, ln1, bufA, DM);

  // 2) q/k/v projections (relu on q,k)
  gemm_wmma_f32<<<blocks512, 256, 0, stream>>>(bufA, Wq, nullptr, nullptr, qbuf, BSR, HID, DM, 1);
  gemm_wmma_f32<<<blocks512, 256, 0, stream>>>(bufA, Wk, nullptr, nullptr, kbuf, BSR, HID, DM, 1);
  gemm_wmma_f32<<<blocks512, 256, 0, stream>>>(bufA, Wv, nullptr, nullptr, vbuf, BSR, HID, DM, 0);

  // 3) linear-attention scan + final carry
  scan_kernel<<<NB * NH, 256, 0, stream>>>(qbuf, kbuf, vbuf, mask, carry, abuf, out_carry);

  // 4) attn = rmsnorm(attn, attn_ln)
  rmsnorm_kernel<<<BSR, 256, 0, stream>>>(abuf, attn_ln, bufA, HID);

  // 5) x1 = attn_norm @ Wo + bo + inputs (skip)
  gemm_wmma_f32<<<blocks512, 256, 0, stream>>>(bufA, Wo, bo, inputs, x1, BSR, DM, HID, 0);

  // 6) x2 = rmsnorm(x1, ln2)
  rmsnorm_kernel<<<BSR, 256, 0, stream>>>(x1, ln2, bufA, DM);

  // 7) gate/up projections
  gemm_wmma_f32<<<blocksFFN, 256, 0, stream>>>(bufA, Wg, nullptr, nullptr, gbuf, BSR, FFN, DM, 0);
  gemm_wmma_f32<<<blocksFFN, 256, 0, stream>>>(bufA, Wu, nullptr, nullptr, ubuf, BSR, FFN, DM, 0);

  // 8) h = silu(g) * u  (in place into gbuf)
  silu_mul_kernel<<<(BSR * FFN + 255) / 256, 256, 0, stream>>>(gbuf, ubuf, BSR * FFN);

  // 9) out = h @ Wd + x1 (skip2)
  gemm_wmma_f32<<<blocks512, 256, 0, stream>>>(gbuf, Wd, nullptr, x1, out_x, BSR, DM, FFN, 0);
}